// MessagePassingLayer_60181081752046
// MI455X (gfx1250) — compile-verified
//
#include <hip/hip_runtime.h>
#include <math.h>

// Problem constants (B,N,D) = (8,200,256)
#define BB   8
#define NN   200
#define DD   256
#define ROWS (BB * NN)          // 1600 token rows
#define K3D  (3 * DD)           // 768

typedef __attribute__((ext_vector_type(2))) float v2f;
typedef __attribute__((ext_vector_type(8))) float v8f;

// ---------------------------------------------------------------------------
// LayerNorm over last dim (D=256). One block (256 threads) per row.
// ---------------------------------------------------------------------------
__global__ void mp_ln_kernel(const float* __restrict__ x,
                             const float* __restrict__ gamma,
                             const float* __restrict__ beta,
                             float* __restrict__ y) {
    __shared__ float s1[DD];
    __shared__ float s2[DD];
    const int row = blockIdx.x;
    const int t   = threadIdx.x;
    const float v = x[(size_t)row * DD + t];
    s1[t] = v;
    s2[t] = v * v;
    __syncthreads();
    for (int s = DD / 2; s > 0; s >>= 1) {
        if (t < s) { s1[t] += s1[t + s]; s2[t] += s2[t + s]; }
        __syncthreads();
    }
    const float mu  = s1[0] * (1.0f / DD);
    const float var = s2[0] * (1.0f / DD) - mu * mu;
    const float rs  = rsqrtf(var + 1e-5f);
    y[(size_t)row * DD + t] = (v - mu) * rs * gamma[t] + beta[t];
}

// ---------------------------------------------------------------------------
// fp32 WMMA GEMM:  C[m,n] = sum_k A[m,k] * W[n*ldw + k]  (+ bias[n])
// A is ROWS x 256 (lda = 256 fixed), K = 256 fixed.
// blockDim = 128 (4 waves); each wave computes one 16x16 tile via
// V_WMMA_F32_16X16X4_F32 (exact fp32 accumulate).
//   grid = (M/16, N/64)
// ---------------------------------------------------------------------------
__global__ void mp_wmma_gemm_k256(const float* __restrict__ A,
                                  const float* __restrict__ W, int ldw,
                                  const float* __restrict__ bias,
                                  float* __restrict__ C, int ldc) {
    const int wave = threadIdx.x >> 5;          // 0..3
    const int lane = threadIdx.x & 31;          // 0..31
    const int m0   = blockIdx.x * 16;
    const int n0   = (blockIdx.y * 4 + wave) * 16;
    const int hv   = lane >> 4;                 // 0 = lanes 0-15, 1 = lanes 16-31
    const int r    = lane & 15;

    // A fragment rows: lane r carries row m0+r; K pair offset 2*hv within k-step.
    const float* __restrict__ Arow = A + (size_t)(m0 + r) * DD;
    // B = W^T tile: B[kk][nn] = W[(n0+nn)*ldw + k0+kk]; lane r supplies column n0+r.
    const float* __restrict__ Wrow = W + (size_t)(n0 + r) * ldw;

    v8f acc = {};
#pragma unroll 8
    for (int k0 = 0; k0 < DD; k0 += 4) {
        const int ka = k0 + 2 * hv;
        v2f a = *(const v2f*)(Arow + ka);        // {A[m][ka], A[m][ka+1]}
        v2f b = *(const v2f*)(Wrow + ka);        // {B[ka][n], B[ka+1][n]}
        acc = __builtin_amdgcn_wmma_f32_16x16x4_f32(
            /*neg_a=*/false, a, /*neg_b=*/false, b,
            /*c_mod=*/(short)0, acc, /*reuse_a=*/false, /*reuse_b=*/false);
    }

    // D layout (32-bit 16x16): VGPR v -> row m0 + v + 8*hv, col n0 + r.
    const int   col = n0 + r;
    const float bv  = bias ? bias[col] : 0.0f;
#pragma unroll
    for (int v = 0; v < 8; ++v) {
        const int row = m0 + v + 8 * hv;
        C[(size_t)row * ldc + col] = acc[v] + bv;
    }
}

// ---------------------------------------------------------------------------
// Dominant kernel: hsum[b,i,k] = sum_j mask[b,i,j] * gelu(ai[b,i,k]+bj[b,j,k]+b1[k])
// cnt[b,i] = sum_j mask[b,i,j].  One block (256 threads) per (b,i) row.
// adj reads are uniform per block -> scalar loads; empty-mask j's skip as a wave.
// Exact gelu: 0.5*u*(1+erf(u/sqrt(2))).
// ---------------------------------------------------------------------------
__global__ void mp_gelu_masksum(const float* __restrict__ AI,
                                const float* __restrict__ BJ,
                                const int* __restrict__ adj,
                                const float* __restrict__ b1,
                                float* __restrict__ hsum,
                                float* __restrict__ cnt) {
    const int row = blockIdx.x;           // b*NN + i
    const int b   = row / NN;
    const int i   = row - b * NN;
    const int t   = threadIdx.x;

    const float aik = AI[(size_t)row * DD + t] + b1[t];
    const int* __restrict__ arow = adj + ((size_t)b * NN + i) * NN;
    const float* __restrict__ bjb = BJ + (size_t)b * NN * DD;

    float acc = 0.0f;
    int   c   = 0;
    for (int j = 0; j < NN; ++j) {
        if (arow[j] != 0) {               // uniform scalar branch
            const float u = aik + bjb[(size_t)j * DD + t];
            acc += 0.5f * u * (1.0f + erff(u * 0.70710678118654752440f));
            ++c;
        }
    }
    hsum[(size_t)row * DD + t] = acc;
    if (t == 0) cnt[row] = (float)c;
}

// ---------------------------------------------------------------------------
// messages = cnt>0 ? msum/cnt + b2 : 0
// (msum here excludes the +cnt*b2 term; folding it in gives exactly msum_total/cnt + b2)
// ---------------------------------------------------------------------------
__global__ void mp_messages(const float* __restrict__ msum,
                            const float* __restrict__ cnt,
                            const float* __restrict__ b2,
                            float* __restrict__ msgs) {
    const int row = blockIdx.x;
    const int t   = threadIdx.x;
    const float c = cnt[row];
    const size_t idx = (size_t)row * DD + t;
    msgs[idx] = (c > 0.0f) ? (msum[idx] / c + b2[t]) : 0.0f;
}

// ---------------------------------------------------------------------------
// GRU gate fusion + residual: out = x + ((1-z)*n + z*x)
// ---------------------------------------------------------------------------
__global__ void mp_gru(const float* __restrict__ x,
                       const float* __restrict__ gi,
                       const float* __restrict__ gh,
                       float* __restrict__ out) {
    const int row = blockIdx.x;
    const int t   = threadIdx.x;
    const size_t g = (size_t)row * K3D + t;

    const float i_r = gi[g],            h_r = gh[g];
    const float i_z = gi[g + DD],       h_z = gh[g + DD];
    const float i_n = gi[g + 2 * DD],   h_n = gh[g + 2 * DD];

    const float r = 1.0f / (1.0f + expf(-(i_r + h_r)));
    const float z = 1.0f / (1.0f + expf(-(i_z + h_z)));
    const float n = tanhf(i_n + r * h_n);

    const float xv = x[(size_t)row * DD + t];
    out[(size_t)row * DD + t] = xv + ((1.0f - z) * n + z * xv);
}

// ---------------------------------------------------------------------------
extern "C" void kernel_launch(void* const* d_in, const int* in_sizes, int n_in,
                              void* d_out, int out_size, void* d_ws, size_t ws_size,
                              hipStream_t stream) {
    const float* x     = (const float*)d_in[0];
    const int*   adj   = (const int*)  d_in[1];
    const float* gamma = (const float*)d_in[2];
    const float* beta  = (const float*)d_in[3];
    const float* W1    = (const float*)d_in[4];   // (D, 2D) row-major
    const float* b1    = (const float*)d_in[5];
    const float* W2    = (const float*)d_in[6];   // (D, D)
    const float* b2    = (const float*)d_in[7];
    const float* W_ih  = (const float*)d_in[8];   // (3D, D)
    const float* W_hh  = (const float*)d_in[9];   // (3D, D)
    const float* b_ih  = (const float*)d_in[10];
    const float* b_hh  = (const float*)d_in[11];
    float* out = (float*)d_out;

    // Workspace layout (floats): ~18.8 MB total.
    float* ws    = (float*)d_ws;
    float* xnorm = ws;                          // ROWS*DD
    float* AI    = xnorm + (size_t)ROWS * DD;   // ROWS*DD
    float* BJ    = AI    + (size_t)ROWS * DD;   // ROWS*DD
    float* hsum  = BJ    + (size_t)ROWS * DD;   // ROWS*DD
    float* msum  = hsum  + (size_t)ROWS * DD;   // ROWS*DD
    float* msgs  = msum  + (size_t)ROWS * DD;   // ROWS*DD
    float* gi    = msgs  + (size_t)ROWS * DD;   // ROWS*3D
    float* gh    = gi    + (size_t)ROWS * K3D;  // ROWS*3D
    float* cnt   = gh    + (size_t)ROWS * K3D;  // ROWS

    const dim3 gblk(128);

    // 1) LayerNorm
    mp_ln_kernel<<<ROWS, DD, 0, stream>>>(x, gamma, beta, xnorm);

    // 2) ai = xnorm @ W1[:, :D]^T ; bj = xnorm @ W1[:, D:]^T   (ldw = 2D)
    mp_wmma_gemm_k256<<<dim3(ROWS / 16, DD / 64), gblk, 0, stream>>>(
        xnorm, W1,       2 * DD, nullptr, AI, DD);
    mp_wmma_gemm_k256<<<dim3(ROWS / 16, DD / 64), gblk, 0, stream>>>(
        xnorm, W1 + DD,  2 * DD, nullptr, BJ, DD);

    // 3) hsum[b,i,:] = sum_j mask * gelu(ai + bj + b1);  cnt = row degree
    mp_gelu_masksum<<<ROWS, DD, 0, stream>>>(AI, BJ, adj, b1, hsum, cnt);

    // 4) msum = hsum @ W2^T ; messages = cnt>0 ? msum/cnt + b2 : 0
    mp_wmma_gemm_k256<<<dim3(ROWS / 16, DD / 64), gblk, 0, stream>>>(
        hsum, W2, DD, nullptr, msum, DD);
    mp_messages<<<ROWS, DD, 0, stream>>>(msum, cnt, b2, msgs);

    // 5) gi = messages @ W_ih^T + b_ih ; gh = x @ W_hh^T + b_hh   (N = 3D)
    mp_wmma_gemm_k256<<<dim3(ROWS / 16, K3D / 64), gblk, 0, stream>>>(
        msgs, W_ih, DD, b_ih, gi, K3D);
    mp_wmma_gemm_k256<<<dim3(ROWS / 16, K3D / 64), gblk, 0, stream>>>(
        x,    W_hh, DD, b_hh, gh, K3D);

    // 6) GRU gates + residual
    mp_gru<<<ROWS, DD, 0, stream>>>(x, gi, gh, out);
}